// MultiHeadedAttention_25563645346690
// MI455X (gfx1250) — compile-verified
//
#include <hip/hip_runtime.h>
#include <math.h>

#define H   8
#define DH  64
#define DM  512
#define SQ  1024
#define NB  2
#define PCLIP 32
#define QREL_STRIDE 40     // 33 used, padded
#define SROW 1040          // LDS score row stride (1024 + pad)

typedef __attribute__((ext_vector_type(2))) float v2f;
typedef __attribute__((ext_vector_type(8))) float v8f;

__device__ __forceinline__ v8f wmma_f32(v2f a, v2f b, v8f c) {
    // D = A(16x4) x B(4x16) + C(16x16), all fp32  -> v_wmma_f32_16x16x4_f32
    return __builtin_amdgcn_wmma_f32_16x16x4_f32(
        /*neg_a=*/false, a, /*neg_b=*/false, b,
        /*c_mod=*/(short)0, c, /*reuse_a=*/false, /*reuse_b=*/false);
}

#define V8F_ZERO {0.f,0.f,0.f,0.f,0.f,0.f,0.f,0.f}

// ---------------------------------------------------------------- K0: table
__global__ __launch_bounds__(256) void mha_table_kernel(float* __restrict__ table) {
    int i = blockIdx.x * blockDim.x + threadIdx.x;
    if (i >= 65 * 32) return;
    int p  = i >> 5;        // 0..64
    int d2 = i & 31;        // dim pair 0..31
    float div = __expf((float)(2 * d2) * (-0.14391156f)); // -ln(10000)/64
    float ang = (float)p * div;
    table[p * DH + 2 * d2]     = __sinf(ang);
    table[p * DH + 2 * d2 + 1] = __cosf(ang);
}

// ------------------------------------------------- K1: fused QKV projection
// Y = X @ W^T + b, reshaped to [B,H,S,DH]. 1 wave per 16(M)x64(N) tile.
__global__ __launch_bounds__(128) void mha_qkv_kernel(
    const float* __restrict__ query, const float* __restrict__ key,
    const float* __restrict__ value,
    const float* __restrict__ Wq, const float* __restrict__ bq,
    const float* __restrict__ Wk, const float* __restrict__ bk,
    const float* __restrict__ Wv, const float* __restrict__ bv,
    float* __restrict__ qbuf, float* __restrict__ kbuf, float* __restrict__ vbuf) {
    const int job   = blockIdx.x * 4 + (threadIdx.x >> 5);   // 3072 wave-jobs
    const int which = job >> 10;                             // 0=q 1=k 2=v
    const int rem   = job & 1023;
    const int mt    = rem >> 3;                              // 0..127
    const int nt    = rem & 7;                               // 0..7 (== head)
    const int lane  = threadIdx.x & 31;
    const int l16   = lane & 15;
    const int half  = lane >> 4;

    const float* X    = which == 0 ? query : which == 1 ? key : value;
    const float* W    = which == 0 ? Wq    : which == 1 ? Wk  : Wv;
    const float* bias = which == 0 ? bq    : which == 1 ? bk  : bv;
    float*       Out  = which == 0 ? qbuf  : which == 1 ? kbuf : vbuf;

    const int m0 = mt * 16;
    const int n0 = nt * 64;

    v8f acc0 = V8F_ZERO, acc1 = V8F_ZERO, acc2 = V8F_ZERO, acc3 = V8F_ZERO;
    const float* arow = X + (m0 + l16) * DM + 2 * half;
    const float* brow = W + (n0 + l16) * DM + 2 * half;
    #pragma unroll 4
    for (int k0 = 0; k0 < DM; k0 += 4) {
        v2f a  = *(const v2f*)(arow + k0);
        v2f b0 = *(const v2f*)(brow + k0);
        v2f b1 = *(const v2f*)(brow + 16 * DM + k0);
        v2f b2 = *(const v2f*)(brow + 32 * DM + k0);
        v2f b3 = *(const v2f*)(brow + 48 * DM + k0);
        acc0 = wmma_f32(a, b0, acc0);
        acc1 = wmma_f32(a, b1, acc1);
        acc2 = wmma_f32(a, b2, acc2);
        acc3 = wmma_f32(a, b3, acc3);
    }
    #pragma unroll
    for (int r = 0; r < 8; ++r) {
        const int m = m0 + r + 8 * half;
        const int bidx = m >> 10, sidx = m & 1023;
        float* op = Out + ((bidx * H + nt) * SQ + sidx) * DH;
        op[ 0 + l16] = acc0[r] + bias[n0 +  0 + l16];
        op[16 + l16] = acc1[r] + bias[n0 + 16 + l16];
        op[32 + l16] = acc2[r] + bias[n0 + 32 + l16];
        op[48 + l16] = acc3[r] + bias[n0 + 48 + l16];
    }
}

// ------------------------------------------------------- K2: qrel = Q.table^T
// qrel[g][j] = sum_d q[g][d]*table[j][d], g in [0,16384), j in [0,33)
__global__ __launch_bounds__(128) void mha_qrel_kernel(
    const float* __restrict__ qbuf, const float* __restrict__ table,
    float* __restrict__ qrel) {
    const int job = blockIdx.x * 4 + (threadIdx.x >> 5);  // 3072 jobs
    const int mt  = job / 3;
    const int jt  = job % 3;
    const int lane = threadIdx.x & 31;
    const int l16  = lane & 15;
    const int half = lane >> 4;
    const int g0 = mt * 16;
    const int j0 = jt * 16;

    v8f acc = V8F_ZERO;
    const float* qrow = qbuf  + (g0 + l16) * DH + 2 * half;
    const float* trow = table + (j0 + l16) * DH + 2 * half;
    #pragma unroll
    for (int t = 0; t < 16; ++t) {
        v2f a = *(const v2f*)(qrow + 4 * t);
        v2f b = *(const v2f*)(trow + 4 * t);
        acc = wmma_f32(a, b, acc);
    }
    #pragma unroll
    for (int r = 0; r < 8; ++r) {
        const int g = g0 + r + 8 * half;
        const int j = j0 + l16;
        if (j < 33) qrel[g * QREL_STRIDE + j] = acc[r];
    }
}

// ---------------------------------------------------------- K3: attention core
// 1 workgroup (4 waves) per (b,h, 16-row q-tile). Two-pass softmax with the
// unnormalized P tile held in LDS; rel-V via the 33-bucket identity.
__global__ __launch_bounds__(128) void mha_attn_kernel(
    const float* __restrict__ qbuf, const float* __restrict__ kbuf,
    const float* __restrict__ vbuf, const float* __restrict__ qrel,
    const float* __restrict__ table, float* __restrict__ attn_out) {
    extern __shared__ float smem[];
    float* sS   = smem;               // [16][SROW]
    float* sSum = smem + 16 * SROW;   // [16]

    const int qt  = blockIdx.x & 63;
    const int bh  = blockIdx.x >> 6;
    const int q0  = qt * 16;
    const int nkt = qt + 1;
    const int wave = threadIdx.x >> 5;
    const int lane = threadIdx.x & 31;
    const int l16  = lane & 15;
    const int half = lane >> 4;

    const float* Qb = qbuf + (bh * SQ + q0) * DH;
    const float* Kb = kbuf + (size_t)bh * SQ * DH;
    const float* Vb = vbuf + (size_t)bh * SQ * DH;
    const float* QR = qrel + (size_t)(bh * SQ + q0) * QREL_STRIDE;

    // preload the 16x64 Q tile as 16 A-fragments (32 VGPRs)
    v2f aq[16];
    #pragma unroll
    for (int t = 0; t < 16; ++t)
        aq[t] = *(const v2f*)(Qb + l16 * DH + 4 * t + 2 * half);

    // ---- pass 1: scores -> LDS ----
    for (int kt = wave; kt < nkt; kt += 4) {
        const int k0 = kt * 16;
        const float* krow = Kb + (k0 + l16) * DH + 2 * half;
        if (kt + 4 < nkt)
            __builtin_prefetch(Kb + (k0 + 64 + l16) * DH, 0, 1);
        v8f acc = V8F_ZERO;
        #pragma unroll
        for (int t = 0; t < 16; ++t) {
            v2f bf = *(const v2f*)(krow + 4 * t);
            acc = wmma_f32(aq[t], bf, acc);
        }
        #pragma unroll
        for (int r = 0; r < 8; ++r) {
            const int row = r + 8 * half;
            const int q   = q0 + row;
            const int k   = k0 + l16;
            const int dist = k - q;
            int j = dist < -PCLIP ? 0 : dist + PCLIP;
            j = j > PCLIP ? PCLIP : j;
            float sc = (acc[r] + QR[row * QREL_STRIDE + j]) * 0.125f;
            if (k > q) sc = -3.0e38f;
            sS[row * SROW + k] = sc;
        }
    }
    __syncthreads();

    // ---- softmax: wave w owns rows 4w..4w+3; keep P unnormalized in LDS ----
    const int L = q0 + 16;
    #pragma unroll
    for (int rr = 0; rr < 4; ++rr) {
        const int row = wave * 4 + rr;
        float* rowp = sS + row * SROW;
        float m = -3.0e38f;
        for (int c = lane; c < L; c += 32) m = fmaxf(m, rowp[c]);
        #pragma unroll
        for (int off = 16; off; off >>= 1) m = fmaxf(m, __shfl_xor(m, off, 32));
        float sum = 0.f;
        for (int c = lane; c < L; c += 32) {
            float p = __expf(rowp[c] - m);
            rowp[c] = p;
            sum += p;
        }
        #pragma unroll
        for (int off = 16; off; off >>= 1) sum += __shfl_xor(sum, off, 32);
        if (lane == 0) sSum[row] = sum;
    }
    __syncthreads();

    // ---- pass 2: O = P x V ; wave w owns d-subtile 16w..16w+15 ----
    const int d0 = wave * 16;
    v8f oacc = V8F_ZERO;
    for (int kt = 0; kt < nkt; ++kt) {
        const int k0 = kt * 16;
        if (kt + 1 < nkt)
            __builtin_prefetch(Vb + (k0 + 16) * DH + d0 + l16, 0, 1);
        #pragma unroll
        for (int t = 0; t < 4; ++t) {
            v2f a = *(const v2f*)(sS + l16 * SROW + k0 + 4 * t + 2 * half);
            const float* Vp = Vb + (k0 + 4 * t + 2 * half) * DH + d0 + l16;
            v2f bf; bf.x = Vp[0]; bf.y = Vp[DH];
            oacc = wmma_f32(a, bf, oacc);
        }
    }

    // ---- epilogue: normalize, rel-V buckets, store ----
    const int bb = bh >> 3;
    const int hh = bh & 7;
    #pragma unroll
    for (int r = 0; r < 8; ++r) {
        const int row = r + 8 * half;
        const int qg  = q0 + row;
        const int d   = d0 + l16;
        const float inv = 1.0f / sSum[row];
        float o = oacc[r] * inv;
        const int kstart = qg >= 31 ? qg - 31 : 0;
        float pmass = 0.f, rel = 0.f;
        for (int k = kstart; k <= qg; ++k) {
            float p = sS[row * SROW + k] * inv;     // buckets 1..32: single k each
            pmass += p;
            rel = fmaf(p, table[(k - qg + PCLIP) * DH + d], rel);
        }
        rel = fmaf(1.0f - pmass, table[d], rel);    // bucket 0 (all clipped k)
        attn_out[(bb * SQ + qg) * DM + hh * DH + d] = o + rel;
    }
}

// ----------------------------------------- K4a: output projection + residual
__global__ __launch_bounds__(128) void mha_oproj_kernel(
    const float* __restrict__ attn, const float* __restrict__ Wo,
    const float* __restrict__ bo, const float* __restrict__ query,
    float* __restrict__ xbuf) {
    const int job = blockIdx.x * 4 + (threadIdx.x >> 5);  // 1024 jobs
    const int mt = job >> 3;
    const int nt = job & 7;
    const int lane = threadIdx.x & 31;
    const int l16  = lane & 15;
    const int half = lane >> 4;
    const int m0 = mt * 16;
    const int n0 = nt * 64;

    v8f acc0 = V8F_ZERO, acc1 = V8F_ZERO, acc2 = V8F_ZERO, acc3 = V8F_ZERO;
    const float* arow = attn + (m0 + l16) * DM + 2 * half;
    const float* brow = Wo   + (n0 + l16) * DM + 2 * half;
    #pragma unroll 4
    for (int k0 = 0; k0 < DM; k0 += 4) {
        v2f a  = *(const v2f*)(arow + k0);
        v2f b0 = *(const v2f*)(brow + k0);
        v2f b1 = *(const v2f*)(brow + 16 * DM + k0);
        v2f b2 = *(const v2f*)(brow + 32 * DM + k0);
        v2f b3 = *(const v2f*)(brow + 48 * DM + k0);
        acc0 = wmma_f32(a, b0, acc0);
        acc1 = wmma_f32(a, b1, acc1);
        acc2 = wmma_f32(a, b2, acc2);
        acc3 = wmma_f32(a, b3, acc3);
    }
    #pragma unroll
    for (int r = 0; r < 8; ++r) {
        const int m = m0 + r + 8 * half;
        float* op = xbuf + m * DM + n0;
        const float* rp = query + m * DM + n0;
        op[ 0 + l16] = acc0[r] + bo[n0 +  0 + l16] + rp[ 0 + l16];
        op[16 + l16] = acc1[r] + bo[n0 + 16 + l16] + rp[16 + l16];
        op[32 + l16] = acc2[r] + bo[n0 + 32 + l16] + rp[32 + l16];
        op[48 + l16] = acc3[r] + bo[n0 + 48 + l16] + rp[48 + l16];
    }
}

// -------------------------------------------- K4b: LayerNorm (ddof=1 variant)
__global__ __launch_bounds__(256) void mha_ln_kernel(
    const float* __restrict__ x, const float* __restrict__ w,
    const float* __restrict__ b, float* __restrict__ out) {
    const int row  = blockIdx.x * 8 + (threadIdx.x >> 5);
    const int lane = threadIdx.x & 31;
    const float* xr = x + row * DM;
    float vals[16];
    float s = 0.f;
    #pragma unroll
    for (int i = 0; i < 16; ++i) { vals[i] = xr[lane + 32 * i]; s += vals[i]; }
    #pragma unroll
    for (int off = 16; off; off >>= 1) s += __shfl_xor(s, off, 32);
    const float mean = s * (1.0f / (float)DM);
    float vs = 0.f;
    #pragma unroll
    for (int i = 0; i < 16; ++i) { float d = vals[i] - mean; vs = fmaf(d, d, vs); }
    #pragma unroll
    for (int off = 16; off; off >>= 1) vs += __shfl_xor(vs, off, 32);
    const float stdv = sqrtf(vs / (float)(DM - 1));      // unbiased (ddof=1)
    const float inv  = 1.0f / (stdv + 1e-6f);
    #pragma unroll
    for (int i = 0; i < 16; ++i) {
        const int c = lane + 32 * i;
        out[row * DM + c] = fmaf(w[c], (vals[i] - mean) * inv, b[c]);
    }
}

// ------------------------------------------------------------------ launcher
extern "C" void kernel_launch(void* const* d_in, const int* in_sizes, int n_in,
                              void* d_out, int out_size, void* d_ws, size_t ws_size,
                              hipStream_t stream) {
    const float* query = (const float*)d_in[0];
    const float* key   = (const float*)d_in[1];
    const float* value = (const float*)d_in[2];
    const float* Wq    = (const float*)d_in[3];
    const float* bq    = (const float*)d_in[4];
    const float* Wk    = (const float*)d_in[5];
    const float* bk    = (const float*)d_in[6];
    const float* Wv    = (const float*)d_in[7];
    const float* bv    = (const float*)d_in[8];
    const float* Wo    = (const float*)d_in[9];
    const float* bo    = (const float*)d_in[10];
    const float* ln_w  = (const float*)d_in[11];
    const float* ln_b  = (const float*)d_in[12];

    float* ws    = (float*)d_ws;
    float* table = ws;                               // 65*64        = 4160
    float* qbuf  = table + 4160;                     // 2*8*1024*64  = 1048576
    float* kbuf  = qbuf  + 1048576;
    float* vbuf  = kbuf  + 1048576;
    float* qrel  = vbuf  + 1048576;                  // 16384*40     = 655360
    float* attn  = qrel  + 655360;                   // 2*1024*512   = 1048576
    float* xbuf  = attn  + 1048576;                  // 2*1024*512   = 1048576

    const size_t smem_bytes = (size_t)(16 * SROW + 16) * sizeof(float); // ~65 KB

    mha_table_kernel<<<9, 256, 0, stream>>>(table);
    mha_qkv_kernel<<<768, 128, 0, stream>>>(query, key, value,
                                            Wq, bq, Wk, bk, Wv, bv,
                                            qbuf, kbuf, vbuf);
    mha_qrel_kernel<<<768, 128, 0, stream>>>(qbuf, table, qrel);
    mha_attn_kernel<<<NB * H * (SQ / 16), 128, smem_bytes, stream>>>(
        qbuf, kbuf, vbuf, qrel, table, attn);
    mha_oproj_kernel<<<256, 128, 0, stream>>>(attn, Wo, bo, query, xbuf);
    mha_ln_kernel<<<256, 256, 0, stream>>>(xbuf, ln_w, ln_b, (float*)d_out);
}